// Gfusedmax_73521250173285
// MI455X (gfx1250) — compile-verified
//
#include <hip/hip_runtime.h>
#include <math.h>

// ---------------- problem constants (match reference) ----------------
#define NUM_GRAPHS_C   512
#define GRAPH_SIZE_C   4096
#define EDGES_PER_G    (GRAPH_SIZE_C * 16)   // 65536
#define LAM_C          1.0f
#define N_ITERS_C      20
#define BLOCK_THREADS  1024                  // 32 waves (wave32)

typedef __attribute__((ext_vector_type(2))) float v2f;
typedef __attribute__((ext_vector_type(8))) float v8f;

// ---------------------------------------------------------------------
// 32-lane (one wave32) full sum using V_WMMA_F32_16X16X4_F32.
// A (16x4): VGPR0 lanes0-15 = A[m][0], lanes16-31 = A[m][2]; VGPR1 = 0.
// B (4x16) = ones  =>  D[m][n] = v[m] + v[m+16]  (row-constant).
// D layout: VGPR r holds M=r (lanes0-15) / M=8+r (lanes16-31), so the sum of
// the 8 D registers in one lane covers half the rows; one xor-16 shuffle adds
// the other half -> total sum of all 32 lane values, in every lane.
// Requires EXEC all-ones: call only from fully-active, non-divergent code.
// ---------------------------------------------------------------------
__device__ inline float wave_sum32(float v) {
#if defined(__gfx1250__) && __has_builtin(__builtin_amdgcn_wmma_f32_16x16x4_f32)
    v2f a; a.x = v;    a.y = 0.0f;
    v2f b; b.x = 1.0f; b.y = 1.0f;
    v8f c = {};
    c = __builtin_amdgcn_wmma_f32_16x16x4_f32(
            /*neg_a=*/false, a, /*neg_b=*/false, b,
            /*c_mod=*/(short)0, c, /*reuse_a=*/false, /*reuse_b=*/false);
    float p = c[0] + c[1] + c[2] + c[3] + c[4] + c[5] + c[6] + c[7];
    p += __shfl_xor(p, 16, 32);
    return p;
#else
    for (int off = 16; off > 0; off >>= 1) v += __shfl_xor(v, off, 32);
    return v;
#endif
}

// Block-wide sum (1024 threads = 32 waves). `red` needs >= 32 floats.
// All threads must reach this together (uniform control flow).
__device__ inline float block_sum(float v, float* red, int tid) {
    float w = wave_sum32(v);                 // per-wave total (all lanes)
    if ((tid & 31) == 0) red[tid >> 5] = w;
    __syncthreads();
    float t = red[tid & 31];                 // 32 wave partials
    float total = wave_sum32(t);             // block total (all lanes)
    __syncthreads();                         // red reusable afterwards
    return total;
}

__device__ inline float block_max(float v, float* red, int tid) {
    for (int off = 16; off > 0; off >>= 1) v = fmaxf(v, __shfl_xor(v, off, 32));
    if ((tid & 31) == 0) red[tid >> 5] = v;
    __syncthreads();
    float t = red[tid & 31];
    for (int off = 16; off > 0; off >>= 1) t = fmaxf(t, __shfl_xor(t, off, 32));
    __syncthreads();
    return t;
}

// ---------------- kernel 0: init global degmax scratch ----------------
__global__ void init_scratch_kernel(int* degmax) {
    if (threadIdx.x == 0) *degmax = 0;
}

// ---------------- kernel 1: global max degree (for `step`) ------------
__global__ __launch_bounds__(BLOCK_THREADS)
void degree_kernel(const int2* __restrict__ edges, int* __restrict__ degmax) {
    __shared__ int deg[GRAPH_SIZE_C];
    __shared__ int smax;
    const int g   = blockIdx.x;
    const int tid = threadIdx.x;
    const long long ebase = (long long)g * EDGES_PER_G;
    const int nbase = g * GRAPH_SIZE_C;

    for (int i = tid; i < GRAPH_SIZE_C; i += BLOCK_THREADS) deg[i] = 0;
    if (tid == 0) smax = 0;
    __syncthreads();

    for (int k = tid; k < EDGES_PER_G; k += BLOCK_THREADS) {
        int2 e = edges[ebase + k];
        atomicAdd(&deg[e.x - nbase], 1);
        atomicAdd(&deg[e.y - nbase], 1);
    }
    __syncthreads();

    int m = 0;
    for (int i = tid; i < GRAPH_SIZE_C; i += BLOCK_THREADS) m = max(m, deg[i]);
    atomicMax(&smax, m);
    __syncthreads();
    if (tid == 0) atomicMax(degmax, smax);
}

// ---------------- kernel 2: fused lasso iterations + sparsemax --------
// One workgroup per graph. Entire graph state lives in LDS (304 KB/WGP):
//   u[65536] | x[4096] | dtu[4096] | z[4096] | red[64]
__global__ __launch_bounds__(BLOCK_THREADS)
void gfusedmax_kernel(const float* __restrict__ x,
                      const int2*  __restrict__ edges,
                      const int*   __restrict__ degmax,
                      float*       __restrict__ out) {
    extern __shared__ float smem[];
    float* u_s   = smem;                          // 65536
    float* x_s   = u_s   + EDGES_PER_G;           // 4096
    float* dtu_s = x_s   + GRAPH_SIZE_C;          // 4096
    float* z_s   = dtu_s + GRAPH_SIZE_C;          // 4096
    float* red_s = z_s   + GRAPH_SIZE_C;          // 64

    const int g   = blockIdx.x;
    const int tid = threadIdx.x;
    const long long ebase = (long long)g * EDGES_PER_G;
    const int nbase = g * GRAPH_SIZE_C;

    const float step = 1.0f / (2.0f * fmaxf((float)(*degmax), 1.0f));

    for (int i = tid; i < GRAPH_SIZE_C; i += BLOCK_THREADS) x_s[i] = x[nbase + i];
    for (int k = tid; k < EDGES_PER_G; k += BLOCK_THREADS)  u_s[k] = 0.0f;
    __syncthreads();

    // N_ITERS update passes + 1 final scatter for fused = x - D^T u
    for (int t = 0; t <= N_ITERS_C; ++t) {
        for (int i = tid; i < GRAPH_SIZE_C; i += BLOCK_THREADS) dtu_s[i] = 0.0f;
        __syncthreads();

        for (int k = tid; k < EDGES_PER_G; k += BLOCK_THREADS) {
            if (k + BLOCK_THREADS < EDGES_PER_G)            // uniform branch
                __builtin_prefetch(&edges[ebase + k + BLOCK_THREADS], 0, 1);
            int2 e  = edges[ebase + k];
            float u = u_s[k];
            atomicAdd(&dtu_s[e.x - nbase],  u);
            atomicAdd(&dtu_s[e.y - nbase], -u);
        }
        __syncthreads();

        for (int i = tid; i < GRAPH_SIZE_C; i += BLOCK_THREADS)
            z_s[i] = x_s[i] - dtu_s[i];
        __syncthreads();

        if (t < N_ITERS_C) {
            for (int k = tid; k < EDGES_PER_G; k += BLOCK_THREADS) {
                int2 e = edges[ebase + k];
                float un = u_s[k] + step * (z_s[e.x - nbase] - z_s[e.y - nbase]);
                u_s[k] = fminf(fmaxf(un, -LAM_C), LAM_C);
            }
            __syncthreads();
        }
    }
    // z_s now holds fused (GAMMA == 1.0 so sparsemax input == z_s)

    // ---------------- sparsemax: bisection for tau -------------------
    float m = -INFINITY;
    for (int i = tid; i < GRAPH_SIZE_C; i += BLOCK_THREADS) m = fmaxf(m, z_s[i]);
    const float zmax = block_max(m, red_s, tid);

    // f(tau) = sum(max(z - tau, 0)) : f(zmax)=0, f(zmax-1)>=1
    float lo = zmax - 1.0f, hi = zmax;
    for (int it = 0; it < 36; ++it) {
        float mid = 0.5f * (lo + hi);
        float p = 0.0f;
        for (int i = tid; i < GRAPH_SIZE_C; i += BLOCK_THREADS)
            p += fmaxf(z_s[i] - mid, 0.0f);
        float s = block_sum(p, red_s, tid);   // uniform across block
        if (s >= 1.0f) lo = mid; else hi = mid;
    }

    // exact refinement on the identified support: tau = (sum - 1) / k
    float pk = 0.0f, ps = 0.0f;
    for (int i = tid; i < GRAPH_SIZE_C; i += BLOCK_THREADS) {
        float zi = z_s[i];
        if (zi > lo) { pk += 1.0f; ps += zi; }
    }
    float ksum = block_sum(pk, red_s, tid);
    float ssum = block_sum(ps, red_s, tid);
    float tau  = (ssum - 1.0f) / fmaxf(ksum, 1.0f);

    for (int i = tid; i < GRAPH_SIZE_C; i += BLOCK_THREADS)
        out[nbase + i] = fmaxf(z_s[i] - tau, 0.0f);
}

// ---------------------------------------------------------------------
extern "C" void kernel_launch(void* const* d_in, const int* in_sizes, int n_in,
                              void* d_out, int out_size, void* d_ws, size_t ws_size,
                              hipStream_t stream) {
    const float* x     = (const float*)d_in[0];
    const int2*  edges = (const int2*)d_in[1];
    int num_graphs = in_sizes[0] / GRAPH_SIZE_C;   // 512 for reference sizes
    int* degmax = (int*)d_ws;

    init_scratch_kernel<<<1, 32, 0, stream>>>(degmax);
    degree_kernel<<<num_graphs, BLOCK_THREADS, 0, stream>>>(edges, degmax);

    size_t lds_bytes = (size_t)(EDGES_PER_G + 3 * GRAPH_SIZE_C + 64) * sizeof(float);
    gfusedmax_kernel<<<num_graphs, BLOCK_THREADS, lds_bytes, stream>>>(
        x, edges, degmax, (float*)d_out);
}